// HierarchicalProCoWrapper_49400713838609
// MI455X (gfx1250) — compile-verified
//
#include <hip/hip_runtime.h>

typedef float v2f __attribute__((ext_vector_type(2)));
typedef float v8f __attribute__((ext_vector_type(8)));

#define FEAT   128
#define NNODES 9281
#define NROWS  8192
#define NLEAF  8192
#define LDP    1152      // padded proto-logit leading dim (18*64)
#define TS     68        // LDS tile row stride: 64 + 4 pad -> banks (4r+k) conflict-free

__device__ __forceinline__ v8f wmma_f32_k4(v2f a, v2f b, v8f c) {
  // D = A(16x4 f32) * B(4x16 f32) + C(16x16 f32)
  return __builtin_amdgcn_wmma_f32_16x16x4_f32(
      /*neg_a=*/false, a, /*neg_b=*/false, b,
      /*c_mod=*/(short)0, c, /*reuse_a=*/false, /*reuse_b=*/false);
}

// ---------------------------------------------------------------------------
// Kernel 1: gather leaf columns of W into a dense, col-major (per-leaf rows)
// matrix: Wg_t[j*128 + k] = W[k*9281 + leaf_ids[j]].  Coalesced writes;
// scattered reads hit L2 (W is only 4.75 MB).
// ---------------------------------------------------------------------------
__global__ void gather_wg(const float* __restrict__ W,
                          const int* __restrict__ leaf_ids,
                          float* __restrict__ Wg) {
  int idx = blockIdx.x * blockDim.x + threadIdx.x;   // 0 .. 8192*128-1
  int j = idx >> 7;
  int k = idx & 127;
  Wg[idx] = W[(size_t)k * NNODES + leaf_ids[j]];
}

// ---------------------------------------------------------------------------
// Kernel 2: proto GEMM  P[0:8192, 0:1152) = features @ W[:, 0:1152)
// 64x64 tile per block, 256 threads = 8 waves, each wave: 16-row strip x two
// 16-col subtiles.  K staged in two 64-wide LDS stages.
// ---------------------------------------------------------------------------
__global__ void proto_gemm(const float* __restrict__ F,
                           const float* __restrict__ W,
                           float* __restrict__ P) {
  __shared__ __align__(16) float sA[64 * TS];   // [row][k]
  __shared__ __align__(16) float sB[64 * TS];   // [col][k]

  const int tid   = threadIdx.x;
  const int lane  = tid & 31;
  const int wave  = tid >> 5;
  const int lr    = lane & 15;
  const int kh    = (lane >> 4) << 1;   // lane-half K offset: 0 or 2
  const int hi8   = (lane >> 4) << 3;   // C-matrix row offset: 0 or 8
  const int strip = wave & 3;           // 16-row strip within 64-row tile
  const int cb    = (wave >> 2) << 5;   // 32-col half within 64-col tile
  const int r0    = blockIdx.x * 64;
  const int n0    = blockIdx.y * 64;

  v8f acc0 = {}; v8f acc1 = {};

  for (int ks = 0; ks < FEAT; ks += 64) {
    __syncthreads();
    // Stage A tile: 64 rows x 64 K, float4 coalesced.
#pragma unroll
    for (int i = 0; i < 4; ++i) {
      int q  = tid + 256 * i;           // 0..1023 float4s
      int r  = q >> 4;
      int c4 = (q & 15) << 2;
      *(float4*)(&sA[r * TS + c4]) =
          *(const float4*)(F + (size_t)(r0 + r) * FEAT + ks + c4);
    }
    // Stage B tile: 64 cols x 64 K from W (row-major, ld=9281).
    // Scalar loads coalesced over n (odd ld forbids aligned float4).
#pragma unroll
    for (int i = 0; i < 16; ++i) {
      int q = tid + 256 * i;            // 0..4095
      int k = q >> 6;
      int n = q & 63;
      sB[n * TS + k] = W[(size_t)(ks + k) * NNODES + n0 + n];
    }
    __syncthreads();
#pragma unroll
    for (int k = 0; k < 64; k += 4) {
      v2f a  = *(const v2f*)(&sA[(strip * 16 + lr) * TS + k + kh]);
      v2f b0 = *(const v2f*)(&sB[(cb      + lr) * TS + k + kh]);
      v2f b1 = *(const v2f*)(&sB[(cb + 16 + lr) * TS + k + kh]);
      acc0 = wmma_f32_k4(a, b0, acc0);
      acc1 = wmma_f32_k4(a, b1, acc1);
    }
  }

  const int iBase = r0 + strip * 16 + hi8;
#pragma unroll
  for (int v = 0; v < 8; ++v) {
    int i = iBase + v;
    P[(size_t)i * LDP + n0 + cb + lr]      = acc0[v];
    P[(size_t)i * LDP + n0 + cb + 16 + lr] = acc1[v];
  }
}

// ---------------------------------------------------------------------------
// Kernel 3: main GEMM (features @ Wg_t) with fused hierarchical epilogue:
// out[i,j] = P[i,0] + max(P[i,p0[j]], P[i,p1[j]]) + dot(F[i,:], Wg[j,:])
// ---------------------------------------------------------------------------
__global__ void leaf_gemm(const float* __restrict__ F,
                          const float* __restrict__ Wg,
                          const float* __restrict__ P,
                          const int* __restrict__ paths,
                          float* __restrict__ out) {
  __shared__ __align__(16) float sA[64 * TS];   // [row][k]
  __shared__ __align__(16) float sB[64 * TS];   // [col][k]

  const int tid   = threadIdx.x;
  const int lane  = tid & 31;
  const int wave  = tid >> 5;
  const int lr    = lane & 15;
  const int kh    = (lane >> 4) << 1;
  const int hi8   = (lane >> 4) << 3;
  const int strip = wave & 3;
  const int cb    = (wave >> 2) << 5;
  const int r0    = blockIdx.x * 64;
  const int j0    = blockIdx.y * 64;

  v8f acc0 = {}; v8f acc1 = {};

  for (int ks = 0; ks < FEAT; ks += 64) {
    __syncthreads();
#pragma unroll
    for (int i = 0; i < 4; ++i) {
      int q  = tid + 256 * i;
      int r  = q >> 4;
      int c4 = (q & 15) << 2;
      *(float4*)(&sA[r * TS + c4]) =
          *(const float4*)(F + (size_t)(r0 + r) * FEAT + ks + c4);
    }
    // B tile from Wg_t (col-major gathered weights, ld=128): float4 coalesced.
#pragma unroll
    for (int i = 0; i < 4; ++i) {
      int q  = tid + 256 * i;
      int n  = q >> 4;
      int c4 = (q & 15) << 2;
      *(float4*)(&sB[n * TS + c4]) =
          *(const float4*)(Wg + (size_t)(j0 + n) * FEAT + ks + c4);
    }
    __syncthreads();
#pragma unroll
    for (int k = 0; k < 64; k += 4) {
      v2f a  = *(const v2f*)(&sA[(strip * 16 + lr) * TS + k + kh]);
      v2f b0 = *(const v2f*)(&sB[(cb      + lr) * TS + k + kh]);
      v2f b1 = *(const v2f*)(&sB[(cb + 16 + lr) * TS + k + kh]);
      acc0 = wmma_f32_k4(a, b0, acc0);
      acc1 = wmma_f32_k4(a, b1, acc1);
    }
  }

  // Fused epilogue: root + best-proto + leaf logit, streamed to out.
  const int iBase = r0 + strip * 16 + hi8;
  const int jA = j0 + cb + lr;
  const int jB = jA + 16;
  const int2 ppA = ((const int2*)paths)[jA];   // proto_paths is (NLEAF, 2) i32
  const int2 ppB = ((const int2*)paths)[jB];
#pragma unroll
  for (int v = 0; v < 8; ++v) {
    int i = iBase + v;
    const float* Pr = P + (size_t)i * LDP;
    float root = Pr[0];
    out[(size_t)i * NLEAF + jA] = root + fmaxf(Pr[ppA.x], Pr[ppA.y]) + acc0[v];
    out[(size_t)i * NLEAF + jB] = root + fmaxf(Pr[ppB.x], Pr[ppB.y]) + acc1[v];
  }
}

// ---------------------------------------------------------------------------
extern "C" void kernel_launch(void* const* d_in, const int* in_sizes, int n_in,
                              void* d_out, int out_size, void* d_ws, size_t ws_size,
                              hipStream_t stream) {
  (void)in_sizes; (void)n_in; (void)out_size; (void)ws_size;
  const float* F     = (const float*)d_in[0];   // (8192, 128) f32
  const float* W     = (const float*)d_in[1];   // (128, 9281) f32
  const int*   paths = (const int*)d_in[2];     // (8192, 2) i32
  const int*   lids  = (const int*)d_in[3];     // (8192,) i32
  float* out = (float*)d_out;                   // (8192, 8192) f32

  float* Wg = (float*)d_ws;                                             // 4 MB
  float* P  = (float*)((char*)d_ws + (size_t)NLEAF * FEAT * sizeof(float)); // 36 MB

  gather_wg<<<(NLEAF * FEAT) / 256, 256, 0, stream>>>(W, lids, Wg);
  proto_gemm<<<dim3(NROWS / 64, LDP / 64), 256, 0, stream>>>(F, W, P);
  leaf_gemm<<<dim3(NROWS / 64, NLEAF / 64), 256, 0, stream>>>(F, Wg, P, paths, out);
}